// ScaledDotProductAttention_20598663152114
// MI455X (gfx1250) — compile-verified
//
#include <hip/hip_runtime.h>
#include <math.h>

// Problem constants (B=1)
constexpr int H = 16;
constexpr int S = 4096;
constexpr int D = 64;
constexpr float SCALE = 0.125f; // 1/sqrt(64)

typedef float v2f __attribute__((ext_vector_type(2)));
typedef float v8f __attribute__((ext_vector_type(8)));

// LDS layout (floats): scores[16][4096] | red[16][16] | rowmax[16] | rowsum[16]
constexpr int SC_FLOATS   = 16 * S;
constexpr int RED_FLOATS  = 16 * 16;
constexpr int SMEM_FLOATS = SC_FLOATS + RED_FLOATS + 16 + 16;

__global__ __launch_bounds__(256, 1)
void sdpa_fused_f32wmma_kernel(const float* __restrict__ q,
                               const float* __restrict__ k,
                               const float* __restrict__ v,
                               const unsigned char* __restrict__ mask,
                               float* __restrict__ out,
                               float* __restrict__ attn) {
    extern __shared__ float smem[];
    float* sc     = smem;                 // [16][S]
    float* red    = smem + SC_FLOATS;     // [16][16]
    float* rowmax = red + RED_FLOATS;     // [16]
    float* rowsum = rowmax + 16;          // [16]

    const int qt   = blockIdx.x;          // 0..255  (16-row query tile)
    const int h    = blockIdx.y;          // 0..15   (head)
    const int tid  = threadIdx.x;
    const int wave = tid >> 5;            // 0..7
    const int lane = tid & 31;
    const int m    = lane & 15;           // M (A/D row sub-index) or N (B/C col)
    const int hi   = lane >> 4;           // lane half select

    const int qbase = qt * 16;

    // ---------------- Phase 1: scores = scale * Q K^T with mask -> LDS -------
    // Preload this wave's Q A-fragments for all 16 K-steps (D=64, K=4/step).
    // A-layout (16x4 f32): lanes 0-15 M=0..15; vgpr0 holds k-chunk elem 2*hi,
    // vgpr1 holds 2*hi+1.
    float a0[16], a1[16];
    {
        const size_t qrow_base = ((size_t)h * S + (qbase + m)) * D;
        #pragma unroll
        for (int kk = 0; kk < 16; ++kk) {
            a0[kk] = q[qrow_base + 4 * kk + 2 * hi];
            a1[kk] = q[qrow_base + 4 * kk + 2 * hi + 1];
        }
    }

    for (int ct = wave; ct < S / 16; ct += 8) {
        const int kcol = ct * 16 + m; // key index handled by this lane (N)
        const size_t krow_base = ((size_t)h * S + kcol) * D;

        // Prefetch next K tile this wave will touch (gfx1250 global_prefetch_b8)
        if (ct + 8 < S / 16) {
            __builtin_prefetch(&k[((size_t)h * S + (ct + 8) * 16 + m) * D], 0, 0);
        }

        v8f c = {0.f, 0.f, 0.f, 0.f, 0.f, 0.f, 0.f, 0.f};
        #pragma unroll
        for (int kk = 0; kk < 16; ++kk) {
            v2f a = {a0[kk], a1[kk]};
            v2f b = {k[krow_base + 4 * kk + 2 * hi],
                     k[krow_base + 4 * kk + 2 * hi + 1]};
            c = __builtin_amdgcn_wmma_f32_16x16x4_f32(
                    false, a, false, b, (short)0, c, false, false);
        }

        // C/D layout: vgpr r -> M = r + 8*hi, lane -> N = m
        #pragma unroll
        for (int r = 0; r < 8; ++r) {
            const int M   = r + 8 * hi;
            const int col = ct * 16 + m;
            float s = c[r] * SCALE;
            const unsigned char mk = mask[(size_t)(qbase + M) * S + col];
            if (mk) s = -INFINITY;
            sc[M * S + col] = s;
        }
    }
    __syncthreads();

    // ---------------- Phase 2: row softmax in LDS, write attn once ----------
    // 16 threads per row: row r = tid/16, sub-lane j = tid%16.
    const int r  = tid >> 4;
    const int j  = tid & 15;
    float mx = -INFINITY;
    for (int c = j; c < S; c += 16) mx = fmaxf(mx, sc[r * S + c]);
    red[r * 16 + j] = mx;
    __syncthreads();
    if (j == 0) {
        float mm = red[r * 16];
        #pragma unroll
        for (int t = 1; t < 16; ++t) mm = fmaxf(mm, red[r * 16 + t]);
        rowmax[r] = mm;
    }
    __syncthreads();
    const float mm = rowmax[r];
    float sum = 0.f;
    for (int c = j; c < S; c += 16) {
        const float e = __expf(sc[r * S + c] - mm);
        sc[r * S + c] = e;
        sum += e;
    }
    red[r * 16 + j] = sum;
    __syncthreads();
    if (j == 0) {
        float ss = red[r * 16];
        #pragma unroll
        for (int t = 1; t < 16; ++t) ss += red[r * 16 + t];
        rowsum[r] = ss;
    }
    __syncthreads();
    const float inv = 1.0f / rowsum[r];
    {
        const size_t arow = ((size_t)h * S + (qbase + r)) * (size_t)S;
        for (int c = j; c < S; c += 16) {
            const float p = sc[r * S + c] * inv;
            sc[r * S + c] = p;
            attn[arow + c] = p;   // the 1.07 GB stream: written exactly once
        }
    }
    __syncthreads();

    // ---------------- Phase 3: out = P @ V (16x4096 @ 4096x64) --------------
    // Waves 0..3 each own one 16-wide output column tile; K swept 4 at a time.
    if (wave < 4) {
        const int nt = wave;
        v8f c2 = {0.f, 0.f, 0.f, 0.f, 0.f, 0.f, 0.f, 0.f};
        for (int kk = 0; kk < S / 4; ++kk) {
            // A = P fragment from LDS (ds_load): lane row = m
            v2f a = {sc[m * S + 4 * kk + 2 * hi],
                     sc[m * S + 4 * kk + 2 * hi + 1]};
            // B = V fragment (rows = K dim, cols = head-dim slice), L2-resident
            const size_t vb = ((size_t)h * S + 4 * kk + 2 * hi) * D + nt * 16 + m;
            v2f b = {v[vb], v[vb + D]};
            c2 = __builtin_amdgcn_wmma_f32_16x16x4_f32(
                     false, a, false, b, (short)0, c2, false, false);
        }
        #pragma unroll
        for (int rr = 0; rr < 8; ++rr) {
            const int M = rr + 8 * hi;
            out[((size_t)h * S + qbase + M) * D + nt * 16 + m] = c2[rr];
        }
    }
}

extern "C" void kernel_launch(void* const* d_in, const int* in_sizes, int n_in,
                              void* d_out, int out_size, void* d_ws, size_t ws_size,
                              hipStream_t stream) {
    (void)in_sizes; (void)n_in; (void)d_ws; (void)ws_size; (void)out_size;
    const float* q = (const float*)d_in[0];
    const float* k = (const float*)d_in[1];
    const float* v = (const float*)d_in[2];
    const unsigned char* mask = (const unsigned char*)d_in[3]; // jnp.bool_ -> u8

    float* out  = (float*)d_out;                       // [1,16,4096,64]
    float* attn = out + (size_t)H * S * D;             // [1,16,4096,4096]

    const dim3 grid(S / 16, H);   // 256 x 16 workgroups
    const dim3 block(256);        // 8 waves (wave32)
    const size_t smem_bytes = (size_t)SMEM_FLOATS * sizeof(float); // ~257 KB

    sdpa_fused_f32wmma_kernel<<<grid, block, smem_bytes, stream>>>(
        q, k, v, mask, out, attn);
}